// tri_LL_38328288150153
// MI455X (gfx1250) — compile-verified
//
#include <hip/hip_runtime.h>

#define D   63
#define D2  3969
#define D3  250047

// workspace layout (floats)
#define F_OFF   0
#define C_OFF   250048
#define M_OFF   (C_OFF + D2)
#define WT_OFF  (M_OFF + D2)   // 64x64 padded W^T ; end ~1.03 MB

typedef float v2f __attribute__((ext_vector_type(2)));
typedef float v8f __attribute__((ext_vector_type(8)));

// ---------------------------------------------------------------------------
// Gell-Mann matrices of su(8), sparse representation (<= 8 nonzeros each).
// Ordering matches the reference: 56 off-diagonal (sym/antisym pairs), then
// 7 diagonal ones.
// ---------------------------------------------------------------------------
__device__ __forceinline__ int lm_fill(int a, int* p, int* q, float* re, float* im) {
    if (a < 56) {
        int pp = a >> 1;
        int j = 0;
        while (pp >= 7 - j) { pp -= 7 - j; ++j; }
        int k = j + 1 + pp;
        if ((a & 1) == 0) {
            p[0] = j; q[0] = k; re[0] = 1.f;  im[0] = 0.f;
            p[1] = k; q[1] = j; re[1] = 1.f;  im[1] = 0.f;
        } else {
            p[0] = j; q[0] = k; re[0] = 0.f;  im[0] = -1.f;
            p[1] = k; q[1] = j; re[1] = 0.f;  im[1] = 1.f;
        }
        return 2;
    }
    int l = a - 55;                       // 1..7
    float cc = sqrtf(2.0f / (float)(l * (l + 1)));
    int n = 0;
    for (int t = 0; t < l; ++t) { p[n] = t; q[n] = t; re[n] = cc; im[n] = 0.f; ++n; }
    p[n] = l; q[n] = l; re[n] = -(float)l * cc; im[n] = 0.f; ++n;
    return n;
}

// f_abc = Tr([L_a,L_b] L_c) / (4i) = Im(Tr(L_a L_b L_c)) / 2   (hermitian L)
__global__ __launch_bounds__(256) void fconst_kernel(float* __restrict__ fg) {
    int idx = blockIdx.x * 256 + threadIdx.x;
    if (idx >= D3) return;
    int a = idx / D2;
    int rem = idx - a * D2;
    int b = rem / D;
    int c = rem - b * D;

    int pa[8], qa[8], pb[8], qb[8], pc[8], qc[8];
    float ar[8], ai[8], br[8], bi[8], cr[8], ci[8];
    int na = lm_fill(a, pa, qa, ar, ai);
    int nb = lm_fill(b, pb, qb, br, bi);
    int nc = lm_fill(c, pc, qc, cr, ci);

    float tim = 0.f;
    for (int s = 0; s < na; ++s) {
        for (int t = 0; t < nb; ++t) {
            if (qa[s] != pb[t]) continue;
            float mre = ar[s] * br[t] - ai[s] * bi[t];
            float mim = ar[s] * bi[t] + ai[s] * br[t];
            for (int u = 0; u < nc; ++u) {
                if (pc[u] == qb[t] && qc[u] == pa[s])
                    tim += mre * ci[u] + mim * cr[u];
            }
        }
    }
    fg[idx] = 0.5f * tim;
}

// c = N * (v^T v) / tr(v^T v);  v: diag from r_x, strict upper from r_y
__global__ __launch_bounds__(256) void cmat_kernel(const float* __restrict__ rx,
                                                   const float* __restrict__ ry,
                                                   const float* __restrict__ Nn,
                                                   float* __restrict__ cg) {
    __shared__ float sv[D2];
    __shared__ float sc[D2];
    int tid = threadIdx.x;
    for (int e = tid; e < D2; e += 256) {
        int i = e / D, j = e - i * D;
        float v = 0.f;
        if (j > i)       v = ry[i * (D - 1) - (i * (i - 1)) / 2 + (j - i - 1)];
        else if (j == i) v = rx[i * D - (i * (i - 1)) / 2];
        sv[e] = v;
    }
    __syncthreads();
    for (int e = tid; e < D2; e += 256) {
        int i = e / D, j = e - i * D;
        float acc = 0.f;
        for (int k = 0; k < D; ++k) acc += sv[k * D + i] * sv[k * D + j];
        sc[e] = acc;
    }
    __syncthreads();
    float tr = 0.f;
    for (int i = 0; i < D; ++i) tr += sc[i * D + i];
    float scale = Nn[0] / tr;
    for (int e = tid; e < D2; e += 256) cg[e] = sc[e] * scale;
}

// M[m,n] = -4 * sum_{i,k} ( sum_j f[m,j,k] c[j,i] ) * f[n,i,k]   (c symmetric)
__global__ __launch_bounds__(256) void mmat_kernel(const float* __restrict__ fg,
                                                   const float* __restrict__ cg,
                                                   float* __restrict__ Mg) {
    __shared__ float sC[D2];
    __shared__ float sF[D2];
    __shared__ float sG[D2];
    int tid = threadIdx.x;
    int m = blockIdx.x;
    for (int e = tid; e < D2; e += 256) { sC[e] = cg[e]; sF[e] = fg[m * D2 + e]; }
    __syncthreads();
    for (int e = tid; e < D2; e += 256) {
        int i = e / D, k = e - i * D;
        float acc = 0.f;
        for (int j = 0; j < D; ++j) acc += sF[j * D + k] * sC[j * D + i];
        sG[e] = acc;                         // G[i*D+k], same flat order as f[n]
    }
    __syncthreads();
    if (tid < D) {
        const float* fn = fg + tid * D2;
        float acc = 0.f;
        for (int e = 0; e < D2; ++e) acc += sG[e] * fn[e];
        Mg[m * D + tid] = -4.f * acc;
    }
}

// Wt[i*64+j] = 1.01*(i==j) + 0.01*( M[i,j] + 8*sum_k f[k,i,j]*omega[k] ), padded to 64x64
__global__ __launch_bounds__(256) void wmat_kernel(const float* __restrict__ fg,
                                                   const float* __restrict__ Mg,
                                                   const float* __restrict__ omega,
                                                   float* __restrict__ wtg) {
    int tid = threadIdx.x;
    for (int e = tid; e < 4096; e += 256) {
        int i = e >> 6, j = e & 63;
        float val = 0.f;
        if (i < D && j < D) {
            float h = 0.f;
            for (int k = 0; k < D; ++k) h += fg[k * D2 + i * D + j] * omega[k];
            val = 0.01f * (Mg[i * D + j] + 8.f * h);
            if (i == j) val += 1.01f;
        }
        wtg[e] = val;
    }
}

// ---------------------------------------------------------------------------
// Main GEMM: out[b,i] = sum_j x[b,j] * Wt[i][j]  via V_WMMA_F32_16X16X4_F32.
// 8 waves/block, 16 rows/wave -> 128 rows/block. Row pitch 68 in LDS keeps
// fragment reads bank-conflict-free and 8B aligned. Global traffic moves via
// flat b128 loads/stores (a block's 128 rows are contiguous: 8064 floats).
// ---------------------------------------------------------------------------
#define XP 68
#define WP 68

__global__ __launch_bounds__(256) void gemm_kernel(const float* __restrict__ x,
                                                   const float* __restrict__ wt_g,
                                                   float* __restrict__ out,
                                                   int n_rows) {
    __shared__ float xs[128 * XP];
    __shared__ float wt[64 * WP];
    const int tid  = threadIdx.x;
    const int base = blockIdx.x * 128;
    const bool full = (base + 128) <= n_rows;   // uniform per block

    // stage W (4096 floats): b128 global loads, b128 LDS stores (rows of 64
    // never split a float4; LDS addr (r*68 + 4k)*4 is 16B aligned)
    {
        const float4* wg = (const float4*)wt_g;
        for (int e4 = tid; e4 < 1024; e4 += 256) {
            float4 v = wg[e4];
            int g = e4 << 2;
            *(float4*)&wt[(g >> 6) * WP + (g & 63)] = v;
        }
    }
    // explicit zero for the padded k=63 column of xs
    if (tid < 128) xs[tid * XP + 63] = 0.f;

    if (full) {
        // 128 rows are contiguous: flat b128 loads, scatter into padded rows
        const float4* xg = (const float4*)(x + (size_t)base * D);
        for (int e4 = tid; e4 < (128 * D) / 4; e4 += 256) {
            float4 v = xg[e4];
            int g = e4 << 2;
            int r = g / D, c = g - r * D;
            xs[r * XP + c] = v.x; if (++c == D) { c = 0; ++r; }
            xs[r * XP + c] = v.y; if (++c == D) { c = 0; ++r; }
            xs[r * XP + c] = v.z; if (++c == D) { c = 0; ++r; }
            xs[r * XP + c] = v.w;
        }
    } else {
        for (int e = tid; e < 128 * 64; e += 256) {
            int r = e >> 6, c = e & 63;
            float v = 0.f;
            if (c < D && (base + r) < n_rows) v = x[(size_t)(base + r) * D + c];
            xs[r * XP + c] = v;
        }
    }
    __syncthreads();

    const int wv   = tid >> 5;
    const int lane = tid & 31;
    const int half = lane >> 4;
    const int lm   = lane & 15;
    const int row0 = wv * 16;

    v8f acc0 = {0.f}, acc1 = {0.f}, acc2 = {0.f}, acc3 = {0.f};
#pragma unroll
    for (int s = 0; s < 16; ++s) {
        const int k0 = s * 4 + half * 2;
        v2f a  = *(const v2f*)&xs[(row0 + lm) * XP + k0];
        v2f b0 = *(const v2f*)&wt[(0 * 16 + lm) * WP + k0];
        v2f b1 = *(const v2f*)&wt[(1 * 16 + lm) * WP + k0];
        v2f b2 = *(const v2f*)&wt[(2 * 16 + lm) * WP + k0];
        v2f b3 = *(const v2f*)&wt[(3 * 16 + lm) * WP + k0];
        acc0 = __builtin_amdgcn_wmma_f32_16x16x4_f32(false, a, false, b0, (short)0, acc0, false, false);
        acc1 = __builtin_amdgcn_wmma_f32_16x16x4_f32(false, a, false, b1, (short)0, acc1, false, false);
        acc2 = __builtin_amdgcn_wmma_f32_16x16x4_f32(false, a, false, b2, (short)0, acc2, false, false);
        acc3 = __builtin_amdgcn_wmma_f32_16x16x4_f32(false, a, false, b3, (short)0, acc3, false, false);
    }
    __syncthreads();
#pragma unroll
    for (int v = 0; v < 8; ++v) {
        int r = row0 + v + 8 * half;
        xs[r * XP +  0 + lm] = acc0[v];
        xs[r * XP + 16 + lm] = acc1[v];
        xs[r * XP + 32 + lm] = acc2[v];
        xs[r * XP + 48 + lm] = acc3[v];
    }
    __syncthreads();

    if (full) {
        float4* og = (float4*)(out + (size_t)base * D);
        for (int e4 = tid; e4 < (128 * D) / 4; e4 += 256) {
            int g = e4 << 2;
            int r = g / D, c = g - r * D;
            float4 v;
            v.x = xs[r * XP + c]; if (++c == D) { c = 0; ++r; }
            v.y = xs[r * XP + c]; if (++c == D) { c = 0; ++r; }
            v.z = xs[r * XP + c]; if (++c == D) { c = 0; ++r; }
            v.w = xs[r * XP + c];
            og[e4] = v;
        }
    } else {
        for (int e = tid; e < 128 * D; e += 256) {
            int r = e / D, c = e - r * D;
            if (base + r < n_rows) out[(size_t)base * D + e] = xs[r * XP + c];
        }
    }
}

extern "C" void kernel_launch(void* const* d_in, const int* in_sizes, int n_in,
                              void* d_out, int out_size, void* d_ws, size_t ws_size,
                              hipStream_t stream) {
    const float* x  = (const float*)d_in[0];
    const float* rx = (const float*)d_in[1];
    const float* ry = (const float*)d_in[2];
    const float* Nn = (const float*)d_in[3];
    const float* om = (const float*)d_in[4];
    float* ws = (float*)d_ws;
    float* fg = ws + F_OFF;
    float* cg = ws + C_OFF;
    float* Mg = ws + M_OFF;
    float* wt = ws + WT_OFF;
    float* out = (float*)d_out;

    int n_rows = in_sizes[0] / D;

    fconst_kernel<<<(D3 + 255) / 256, 256, 0, stream>>>(fg);
    cmat_kernel<<<1, 256, 0, stream>>>(rx, ry, Nn, cg);
    mmat_kernel<<<D, 256, 0, stream>>>(fg, cg, Mg);
    wmat_kernel<<<1, 256, 0, stream>>>(fg, Mg, om, wt);

    int nblk = (n_rows + 127) / 128;
    gemm_kernel<<<nblk, 256, 0, stream>>>(x, wt, out, n_rows);
}